// G2InvariantFeatureExtractor_43224550867442
// MI455X (gfx1250) — compile-verified
//
#include <hip/hip_runtime.h>
#include <math.h>

// Problem constants (from reference setup_inputs)
#define BB   32
#define SS   8192
#define DD   7
#define FF   55          // feature count
#define FP   56          // padded K for GEMM1 (14 k-steps of 4)
#define HH   128
#define OO   64
#define TILE 64          // rows per block
#define FSTR 57          // sFeat row stride (odd -> conflict free)
#define HSTR 129         // sH row stride (129 % 64 == 1 -> conflict free)

typedef float v2f __attribute__((ext_vector_type(2)));
typedef float v8f __attribute__((ext_vector_type(8)));

__device__ __forceinline__ float dot7(const float* a, const float* b) {
    float s = 0.f;
#pragma unroll
    for (int i = 0; i < 7; ++i) s += a[i] * b[i];
    return s;
}

__device__ __forceinline__ void cross7(const float* x, const float* y, float* c) {
    const int T[7][3] = {{0,1,2},{0,3,4},{0,6,5},{1,3,5},{1,4,6},{2,3,6},{2,5,4}};
#pragma unroll
    for (int i = 0; i < 7; ++i) c[i] = 0.f;
#pragma unroll
    for (int t = 0; t < 7; ++t) {
        const int i = T[t][0], j = T[t][1], k = T[t][2];
        c[k] += x[i]*y[j] - x[j]*y[i];
        c[i] += x[j]*y[k] - x[k]*y[j];
        c[j] += x[k]*y[i] - x[i]*y[k];
    }
}

__device__ __forceinline__ void load7(const float* p, float* v) {
#pragma unroll
    for (int i = 0; i < 7; ++i) v[i] = p[i];
}

__global__ __launch_bounds__(256)
void g2_fused_kernel(const float* __restrict__ X,    // [B,S,7]
                     const float* __restrict__ W1,   // [55,128]
                     const float* __restrict__ B1,   // [128]
                     const float* __restrict__ G,    // [128] gamma
                     const float* __restrict__ Bt,   // [128] beta
                     const float* __restrict__ W2,   // [128,64]
                     const float* __restrict__ B2,   // [64]
                     float* __restrict__ OUT)        // [B,S,64]
{
    __shared__ float sFeat[TILE * FSTR];
    __shared__ float sH[TILE * HSTR];

    const int tid   = threadIdx.x;
    const int tilesPerSeq = SS / TILE;                 // 128
    const int b  = blockIdx.x / tilesPerSeq;
    const int s0 = (blockIdx.x % tilesPerSeq) * TILE;

    // ---------------- Phase 1: build 55 invariant features per row ----------------
    if (tid < TILE) {
        const int r = tid;
        const int s = s0 + r;
        float* fr = &sFeat[r * FSTR];
#pragma unroll
        for (int c = 0; c < FSTR; ++c) fr[c] = 0.f;

        float xi[7];
        load7(X + ((size_t)b * SS + s) * DD, xi);
        const float nx = sqrtf(dot7(xi, xi));
        fr[0] = nx;

        // neighbor index list, reference order: for k in 1..3: left (if valid), right (if valid)
        int nb[6]; int n = 0;
#pragma unroll
        for (int k = 1; k <= 3; ++k) {
            if (s - k >= 0) nb[n++] = s - k;
            if (s + k < SS) nb[n++] = s + k;
        }

        float y[7], z[7], cxy[7];
        load7(X + ((size_t)b * SS + nb[0]) * DD, y);
        for (int j = 0; j < n; ++j) {
            const bool hasz = (j + 1 < n);
            if (hasz) load7(X + ((size_t)b * SS + nb[j + 1]) * DD, z);
            cross7(xi, y, cxy);
            const int base = 1 + 10 * j;
            fr[base + 0] = nx;
            fr[base + 1] = dot7(xi, y);
            fr[base + 2] = sqrtf(dot7(y, y));
            fr[base + 3] = sqrtf(dot7(cxy, cxy));
            if (hasz) {
                fr[base + 4] = dot7(xi, z);
                fr[base + 5] = dot7(y, z);
                fr[base + 6] = sqrtf(dot7(z, z));
                fr[base + 7] = dot7(cxy, z);     // phi3(x,y,z) == cross(x,y) . z
                // fr[base+8], fr[base+9]: phi3(x,x,y)=phi3(y,y,z)=0 (already zeroed)
#pragma unroll
                for (int i = 0; i < 7; ++i) y[i] = z[i];
            }
        }
    }
    __syncthreads();

    // ---------------- Phase 2: GEMM1  [64x56] @ [56x128]  (WMMA f32 16x16x4) -----
    const int wave  = tid >> 5;
    const int lane  = tid & 31;
    const int laneN = lane & 15;
    const int hi    = lane >> 4;          // 0: K pair (0,1), 1: K pair (2,3)
    const int n0    = wave * 16;          // this wave's H-column tile

    v8f acc[4];
#pragma unroll
    for (int mt = 0; mt < 4; ++mt) acc[mt] = (v8f){0.f,0.f,0.f,0.f,0.f,0.f,0.f,0.f};

    for (int kt = 0; kt < FP / 4; ++kt) {            // 14 k-steps
        const int kA = kt * 4 + 2 * hi;
        v2f bf;
        bf[0] = (kA     < FF) ? W1[kA * HH + n0 + laneN]       : 0.f;
        bf[1] = (kA + 1 < FF) ? W1[(kA + 1) * HH + n0 + laneN] : 0.f;
#pragma unroll
        for (int mt = 0; mt < 4; ++mt) {
            const int m = mt * 16 + laneN;
            v2f af;
            af[0] = sFeat[m * FSTR + kA];
            af[1] = sFeat[m * FSTR + kA + 1];
            acc[mt] = __builtin_amdgcn_wmma_f32_16x16x4_f32(
                false, af, false, bf, (short)0, acc[mt], false, false);
        }
    }

    // D tile layout: lane 0-15 -> col n0+laneN, VGPR v -> row mt*16 + v + 8*hi
    {
        const float bias = B1[n0 + laneN];
#pragma unroll
        for (int mt = 0; mt < 4; ++mt)
#pragma unroll
            for (int v = 0; v < 8; ++v)
                sH[(mt * 16 + v + 8 * hi) * HSTR + n0 + laneN] = acc[mt][v] + bias;
    }
    __syncthreads();

    // ---------------- Phase 3: LayerNorm + GELU(erf) per row ---------------------
    if (tid < TILE) {
        float* hr = &sH[tid * HSTR];
        float sum = 0.f, sq = 0.f;
        for (int c = 0; c < HH; ++c) { const float v = hr[c]; sum += v; sq += v * v; }
        const float mu  = sum * (1.f / HH);
        const float var = sq * (1.f / HH) - mu * mu;
        const float rin = rsqrtf(var + 1e-5f);
        for (int c = 0; c < HH; ++c) {
            float v = (hr[c] - mu) * rin * G[c] + Bt[c];
            v = 0.5f * v * (1.f + erff(v * 0.70710678118654752f));
            hr[c] = v;
        }
    }
    __syncthreads();

    // ---------------- Phase 4: GEMM2  [64x128] @ [128x64]  -----------------------
    const int mt2 = wave & 3;             // row tile
    const int ntb = (wave >> 2) * 2;      // first of two col tiles
    v8f acc2[2];
    acc2[0] = (v8f){0.f,0.f,0.f,0.f,0.f,0.f,0.f,0.f};
    acc2[1] = (v8f){0.f,0.f,0.f,0.f,0.f,0.f,0.f,0.f};

    const int mA = mt2 * 16 + laneN;
    for (int kt = 0; kt < HH / 4; ++kt) {            // 32 k-steps
        const int kA = kt * 4 + 2 * hi;
        v2f af;
        af[0] = sH[mA * HSTR + kA];
        af[1] = sH[mA * HSTR + kA + 1];
#pragma unroll
        for (int t = 0; t < 2; ++t) {
            const int nn = (ntb + t) * 16 + laneN;
            v2f bf;
            bf[0] = W2[kA * OO + nn];
            bf[1] = W2[(kA + 1) * OO + nn];
            acc2[t] = __builtin_amdgcn_wmma_f32_16x16x4_f32(
                false, af, false, bf, (short)0, acc2[t], false, false);
        }
    }

    // ---------------- Phase 5: +b2 and store --------------------------------------
    const size_t rowBase = (size_t)b * SS + s0;
#pragma unroll
    for (int t = 0; t < 2; ++t) {
        const int nn = (ntb + t) * 16 + laneN;
        const float bias = B2[nn];
#pragma unroll
        for (int v = 0; v < 8; ++v) {
            const int row = mt2 * 16 + v + 8 * hi;
            OUT[(rowBase + row) * OO + nn] = acc2[t][v] + bias;
        }
    }
}

extern "C" void kernel_launch(void* const* d_in, const int* in_sizes, int n_in,
                              void* d_out, int out_size, void* d_ws, size_t ws_size,
                              hipStream_t stream) {
    const float* X  = (const float*)d_in[0];
    const float* W1 = (const float*)d_in[1];
    const float* B1 = (const float*)d_in[2];
    const float* G  = (const float*)d_in[3];
    const float* Bt = (const float*)d_in[4];
    const float* W2 = (const float*)d_in[5];
    const float* B2 = (const float*)d_in[6];
    float* OUT = (float*)d_out;

    const int grid = BB * (SS / TILE);   // 32 * 128 = 4096 blocks
    g2_fused_kernel<<<grid, 256, 0, stream>>>(X, W1, B1, G, Bt, W2, B2, OUT);
}